// complex_fuse_base_14156212208090
// MI455X (gfx1250) — compile-verified
//
#include <hip/hip_runtime.h>
#include <hip/hip_bf16.h>
#include <math.h>

#define H_DIM 512
#define D_DIM 256

typedef float v2f __attribute__((ext_vector_type(2)));
typedef float v8f __attribute__((ext_vector_type(8)));

__device__ __forceinline__ float elu_f(float x) {
    return x > 0.0f ? x : expm1f(x);
}

// ---------------------------------------------------------------------------
// Kernel A: fp_n = elu(LayerNorm(drug_fp)) ; one wave32 per row of H=512
// (16 floats per lane, cross-lane reduction via shfl_xor)
// ---------------------------------------------------------------------------
__global__ void ln_elu_rows(const float* __restrict__ in, float* __restrict__ out, int N) {
    const int lane = threadIdx.x & 31;
    const int wid  = (blockIdx.x * blockDim.x + threadIdx.x) >> 5;
    if (wid >= N) return;

    const float4* p = (const float4*)(in + (size_t)wid * H_DIM);
    float4* q = (float4*)(out + (size_t)wid * H_DIM);

    float4 v[4];
    float s = 0.0f, s2 = 0.0f;
#pragma unroll
    for (int j = 0; j < 4; ++j) {
        v[j] = p[j * 32 + lane];                 // coalesced across lanes
        s  += v[j].x + v[j].y + v[j].z + v[j].w;
        s2 += v[j].x * v[j].x + v[j].y * v[j].y + v[j].z * v[j].z + v[j].w * v[j].w;
    }
#pragma unroll
    for (int m = 16; m >= 1; m >>= 1) {
        s  += __shfl_xor(s,  m, 32);
        s2 += __shfl_xor(s2, m, 32);
    }
    const float mean = s  * (1.0f / (float)H_DIM);
    const float var  = s2 * (1.0f / (float)H_DIM) - mean * mean;
    const float inv  = rsqrtf(var + 1e-5f);
#pragma unroll
    for (int j = 0; j < 4; ++j) {
        float4 r;
        r.x = elu_f((v[j].x - mean) * inv);
        r.y = elu_f((v[j].y - mean) * inv);
        r.z = elu_f((v[j].z - mean) * inv);
        r.w = elu_f((v[j].w - mean) * inv);
        q[j * 32 + lane] = r;
    }
}

// ---------------------------------------------------------------------------
// Kernel B: both GEMMs via V_WMMA_F32_16X16X4_F32.
//   blockIdx.z == 0 : x_fp   = elu(fp_n      @ W_fp^T)
//   blockIdx.z == 1 : x_skip =     drug_init @ W_skip^T
// One wave computes one 16x16 f32 tile. A-layout (16x4 f32): lane l holds
// row m=l%16, VGPR j holds K = 2*(l/16)+j  -> contiguous float2 load per lane.
// B-layout (4x16) is the symmetric transpose: lane l holds col n=l%16,
// VGPR j holds K = 2*(l/16)+j -> float2 from W row n. C/D: VGPR v, lane l
// holds C[v + 8*(l/16)][l%16].
// ---------------------------------------------------------------------------
__global__ void gemm_wmma_2(const float* __restrict__ fpn,
                            const float* __restrict__ dinit,
                            const float* __restrict__ Wfp,
                            const float* __restrict__ Wsk,
                            float* __restrict__ xfp,
                            float* __restrict__ xsk,
                            int N) {
    const int lane = threadIdx.x & 31;
    const int wv   = threadIdx.x >> 5;          // 0..3 waves per block
    const int mt   = blockIdx.x;                // M tile
    const int nt   = blockIdx.y * 4 + wv;       // N tile (0..15)
    const bool skip_path = (blockIdx.z != 0);

    const float* A = skip_path ? dinit : fpn;
    const float* W = skip_path ? Wsk   : Wfp;
    float*       C = skip_path ? xsk   : xfp;

    const int hi = lane >> 4;                   // 0 or 1
    const int lo = lane & 15;

    int mrow = mt * 16 + lo;
    if (mrow >= N) mrow = N - 1;                // clamp (N=2000 divides 16; safety)
    const int ncol = nt * 16 + lo;

    const float* arow = A + (size_t)mrow * H_DIM + 2 * hi;
    const float* brow = W + (size_t)ncol * H_DIM + 2 * hi;

    v8f acc = {};
#pragma unroll 4
    for (int k0 = 0; k0 < H_DIM; k0 += 4) {
        v2f a = *(const v2f*)(arow + k0);
        v2f b = *(const v2f*)(brow + k0);
        acc = __builtin_amdgcn_wmma_f32_16x16x4_f32(
            /*neg_a=*/false, a, /*neg_b=*/false, b,
            /*c_mod=*/(short)0, acc, /*reuse_a=*/false, /*reuse_b=*/false);
    }

#pragma unroll
    for (int v = 0; v < 8; ++v) {
        const int row = mt * 16 + v + 8 * hi;
        if (row < N) {
            float val = acc[v];
            if (!skip_path) val = elu_f(val);
            C[(size_t)row * D_DIM + nt * 16 + lo] = val;
        }
    }
}

// ---------------------------------------------------------------------------
// Kernel C: edge stage. One wave32 per edge; each lane owns 8 contiguous
// floats of the D=256 vector. LN reductions via 5 shfl_xor steps.
// ---------------------------------------------------------------------------
__device__ __forceinline__ v8f load8(const float* __restrict__ tab, int row, int lane) {
    return ((const v8f*)(tab + (size_t)row * D_DIM))[lane];
}

__device__ __forceinline__ v8f wave_ln(v8f v) {
    float s = 0.0f, s2 = 0.0f;
#pragma unroll
    for (int j = 0; j < 8; ++j) { s += v[j]; s2 += v[j] * v[j]; }
#pragma unroll
    for (int m = 16; m >= 1; m >>= 1) {
        s  += __shfl_xor(s,  m, 32);
        s2 += __shfl_xor(s2, m, 32);
    }
    const float mean = s  * (1.0f / (float)D_DIM);
    const float var  = s2 * (1.0f / (float)D_DIM) - mean * mean;
    const float inv  = rsqrtf(var + 1e-5f);
#pragma unroll
    for (int j = 0; j < 8; ++j) v[j] = (v[j] - mean) * inv;
    return v;
}

__global__ void edge_kernel(const float* __restrict__ x1,
                            const float* __restrict__ x2,
                            const float* __restrict__ xfp,
                            const float* __restrict__ xsk,
                            const float* __restrict__ wR,
                            const float* __restrict__ wI,
                            const int* __restrict__ idx1,
                            const int* __restrict__ idx2,
                            const int* __restrict__ idx3,
                            float* __restrict__ out, int E) {
    const int lane = threadIdx.x & 31;
    const int e = (blockIdx.x * blockDim.x + threadIdx.x) >> 5;
    if (e >= E) return;

    const int i1 = idx1[e];
    const int i2 = idx2[e];
    const int i3 = idx3[e];

    v8f R1 = wave_ln(load8(x1,  i1, lane) + load8(xfp, i2, lane));
    v8f I1 = wave_ln(load8(x2,  i1, lane) + load8(xsk, i2, lane));
    v8f R2 = wave_ln(load8(xsk, i1, lane) + load8(x2,  i2, lane));
    v8f I2 = wave_ln(load8(xfp, i1, lane) + load8(x1,  i2, lane));

    v8f Rc = wave_ln(R1 * R2 - I1 * I2);
    v8f Ic = wave_ln(R1 * I2 + I1 * R2);

    v8f wr = load8(wR, i3, lane);
    v8f wi = load8(wI, i3, lane);

    float acc = 0.0f;   // r_ + i_ fused: Rc*wr + Ic*wi - Rc*wi + Ic*wr
#pragma unroll
    for (int j = 0; j < 8; ++j) {
        acc += Rc[j] * (wr[j] - wi[j]) + Ic[j] * (wi[j] + wr[j]);
    }
#pragma unroll
    for (int m = 16; m >= 1; m >>= 1) acc += __shfl_xor(acc, m, 32);

    if (lane == 0) out[e] = 1.0f / (1.0f + expf(-acc));
}

// ---------------------------------------------------------------------------
extern "C" void kernel_launch(void* const* d_in, const int* in_sizes, int n_in,
                              void* d_out, int out_size, void* d_ws, size_t ws_size,
                              hipStream_t stream) {
    const float* drug_fp   = (const float*)d_in[0];
    const float* drug_init = (const float*)d_in[1];
    const float* x1        = (const float*)d_in[2];
    const float* x2        = (const float*)d_in[3];
    const float* W_fp      = (const float*)d_in[4];
    const float* W_skip    = (const float*)d_in[5];
    const float* wR        = (const float*)d_in[6];
    const float* wI        = (const float*)d_in[7];
    const int*   idx1      = (const int*)d_in[8];
    const int*   idx2      = (const int*)d_in[9];
    const int*   idx3      = (const int*)d_in[10];
    float* out = (float*)d_out;

    const int N = in_sizes[0] / H_DIM;   // 2000
    const int E = in_sizes[8];           // 200000

    float* fpn = (float*)d_ws;                         // [N, H]
    float* xfp = fpn + (size_t)N * H_DIM;              // [N, D]
    float* xsk = xfp + (size_t)N * D_DIM;              // [N, D]

    // Stage 1: LN + ELU on fingerprint rows (one wave per row, 8 waves/block)
    {
        const int blocks = (N * 32 + 255) / 256;
        ln_elu_rows<<<blocks, 256, 0, stream>>>(drug_fp, fpn, N);
    }
    // Stage 2: both GEMMs via f32 WMMA (4 waves/block, 16x16 tile per wave)
    {
        dim3 grid((N + 15) / 16, 4, 2);
        gemm_wmma_2<<<grid, 128, 0, stream>>>(fpn, drug_init, W_fp, W_skip, xfp, xsk, N);
    }
    // Stage 3: edge stage (one wave per edge, 8 waves/block)
    {
        const long long waves = (long long)E;
        const int blocks = (int)((waves * 32 + 255) / 256);
        edge_kernel<<<blocks, 256, 0, stream>>>(x1, x2, xfp, xsk, wR, wI,
                                                idx1, idx2, idx3, out, E);
    }
}